// MeshConv_13967233646603
// MI455X (gfx1250) — compile-verified
//
#include <hip/hip_runtime.h>
#include <hip/hip_bf16.h>

// ChebNet mesh conv for MI455X (gfx1250, wave32).
// Pipeline: T0 = x (input reused directly, layout [B][M][F]);
//   T1 = L x ; T_k = 2 L T_{k-1} - T_{k-2}  (SpMM via L2-resident atomics)
//   out += A_k @ W_k per Chebyshev order, A_k row-major [200000,64],
//   using V_WMMA_F32_16X16X4_F32 (f32 in / f32 out, matches reference precision).
// This rev packs the W chunk in LDS as fragment-ordered float2 K-pairs so each
// WMMA B operand is a single aligned ds_load_b64 (no v_mov assembly shuffles).

typedef __attribute__((ext_vector_type(2))) float v2f;
typedef __attribute__((ext_vector_type(8))) float v8f;

#define BB    4
#define MM    50000
#define FIN   64
#define KCHEB 6
#define FOUT  64
#define EDG   800000
#define NROWS (BB * MM)            // 200000
#define NT    (NROWS * FIN)        // elements per T buffer (12.8M floats)

// ---------------------------------------------------------------------------
// dst = scale*src (float4-vectorized), or dst = 0 when src == nullptr.
// Used for zero-init of T1 / d_out and for T_k = -T_{k-2} seeding.
// ---------------------------------------------------------------------------
__global__ __launch_bounds__(256) void init_scale_kernel(
    float4* __restrict__ dst, const float4* __restrict__ src,
    float scale, int n4)
{
    int i = blockIdx.x * 256 + threadIdx.x;
    if (i >= n4) return;
    if (src) {
        float4 s = src[i];
        dst[i] = make_float4(scale * s.x, scale * s.y, scale * s.z, scale * s.w);
    } else {
        dst[i] = make_float4(0.f, 0.f, 0.f, 0.f);
    }
}

// ---------------------------------------------------------------------------
// SpMM over COO edges: out[b, row[e], f] += scale*vals[e] * h[b, col[e], f]
// 64 threads per edge, each thread owns 4 contiguous features (float4 gather
// + 4 global_atomic_add_f32). All traffic stays in the 192MB L2 (T = 51MB).
// ---------------------------------------------------------------------------
__global__ __launch_bounds__(256) void cheb_spmm_edges_kernel(
    float* __restrict__ out, const float* __restrict__ h,
    const float* __restrict__ vals, const int* __restrict__ row,
    const int* __restrict__ col, float scale, int nEdges)
{
    int e = blockIdx.x * 4 + (threadIdx.x >> 6);
    if (e >= nEdges) return;
    int c  = (threadIdx.x & 63) * 4;   // column 0..255, step 4
    int b  = c >> 6;                   // batch
    int f  = c & 63;                   // feature (f..f+3 stay inside one batch)
    float v = vals[e] * scale;         // lanes of a wave share e -> one request
    int r  = row[e];
    int cl = col[e];

    const float* hp = h + ((size_t)(b * MM + cl)) * FIN + f;
    float4 hv = *(const float4*)hp;
    float* op = out + ((size_t)(b * MM + r)) * FIN + f;
    atomicAdd(op + 0, v * hv.x);
    atomicAdd(op + 1, v * hv.y);
    atomicAdd(op + 2, v * hv.z);
    atomicAdd(op + 3, v * hv.w);
}

// ---------------------------------------------------------------------------
// GEMM chunk for Chebyshev order k:
//   out[r, fo] += sum_f A[r*64 + f] * W[(f*KCHEB + k)*FOUT + fo]
// One wave computes a 16x64 C tile (4x v8f accumulators).
//
// LDS layouts (fragment-ready):
//  - ldsW2[(q*4 + nt)*16 + n] = { Wk[2q][nt*16+n], Wk[2q+1][nt*16+n] }
//    => B fragment for lane (n = lane&15, kk = (lane>>4)*2) at K-step k0 is
//       one aligned 8B LDS load: ldsW2[(((k0+kk)>>1)*4 + nt)*16 + n].
//  - ldsA: wave-private 16x64 row-major tile, staged coalesced from global;
//    A fragment (K,K+1 pair) is contiguous -> one aligned 8B LDS load.
// Same-wave DS ops are in-order, so no barrier needed for the private A tile.
// ---------------------------------------------------------------------------
__global__ __launch_bounds__(256) void gemm_wmma_chunk_kernel(
    float* __restrict__ out, const float* __restrict__ A,
    const float* __restrict__ W, int kcheb, int nRows)
{
    __shared__ v2f   ldsW2[(FIN / 2) * 4 * 16];   // 2048 float2 = 16 KB
    __shared__ float ldsA[8][16 * FIN];           // 8 waves * 4 KB = 32 KB

    int tid = threadIdx.x;
    // Stage W_k as K-pair fragments: 2048 float2, 8 per thread.
    for (int i = tid; i < (FIN / 2) * 4 * 16; i += 256) {
        int q  = i >> 6;            // K-pair index 0..31
        int fo = i & 63;            // output column 0..63 (nt*16 + n)
        int f0 = q * 2;
        v2f w;
        w.x = W[(f0       * KCHEB + kcheb) * FOUT + fo];
        w.y = W[((f0 + 1) * KCHEB + kcheb) * FOUT + fo];
        ldsW2[i] = w;
    }
    __syncthreads();

    int wave = tid >> 5;
    int lane = tid & 31;
    int row0 = (blockIdx.x * 8 + wave) * 16;
    if (row0 >= nRows) return;                  // no further barriers below

    // Coalesced stage of this wave's 16x64 A tile into LDS.
    const float* Ab = A + (size_t)row0 * FIN;
    float* lA = ldsA[wave];
    #pragma unroll
    for (int it = 0; it < 8; ++it) {
        int idx = it * 128 + lane * 4;
        *(float4*)&lA[idx] = *(const float4*)&Ab[idx];
    }

    // Fragment coordinates (ISA 7.12.2 layouts, wave32):
    // A 16x4 f32: lane&15 = M row; vgpr pair = K,K+1; lane>=16 adds K+2.
    // C 16x16 f32: vgpr i -> row i (+8 for lanes 16..31), col = lane&15.
    int m    = lane & 15;
    int kk   = (lane >> 4) << 1;
    int crow = (lane >> 4) * 8;
    int n    = lane & 15;

    v8f c[4];
    #pragma unroll
    for (int nt = 0; nt < 4; ++nt)
        #pragma unroll
        for (int i = 0; i < 8; ++i)
            c[nt][i] = out[(size_t)(row0 + crow + i) * FOUT + nt * 16 + n];

    #pragma unroll
    for (int k0 = 0; k0 < FIN; k0 += 4) {
        int q = (k0 + kk) >> 1;                          // K-pair index
        v2f a = *(const v2f*)&lA[m * FIN + k0 + kk];     // one ds_load_b64
        #pragma unroll
        for (int nt = 0; nt < 4; ++nt) {
            v2f b = ldsW2[(q * 4 + nt) * 16 + n];        // one ds_load_b64
            c[nt] = __builtin_amdgcn_wmma_f32_16x16x4_f32(
                false, a, false, b, (short)0, c[nt], false, false);
        }
    }

    #pragma unroll
    for (int nt = 0; nt < 4; ++nt)
        #pragma unroll
        for (int i = 0; i < 8; ++i)
            out[(size_t)(row0 + crow + i) * FOUT + nt * 16 + n] = c[nt][i];
}

// ---------------------------------------------------------------------------
extern "C" void kernel_launch(void* const* d_in, const int* in_sizes, int n_in,
                              void* d_out, int out_size, void* d_ws, size_t ws_size,
                              hipStream_t stream)
{
    const float* x    = (const float*)d_in[0];   // [B, M, FIN] == T0 layout
    const float* vals = (const float*)d_in[1];   // [E]
    const int*   row  = (const int*)  d_in[2];   // [E]
    const int*   col  = (const int*)  d_in[3];   // [E]
    const float* W    = (const float*)d_in[4];   // [FIN*K, FOUT]
    float*       out  = (float*)d_out;           // [B, M, FOUT]

    // 3 rotating Chebyshev buffers in workspace (T0 is the input itself).
    float* T[3] = { (float*)d_ws,
                    (float*)d_ws + (size_t)NT,
                    (float*)d_ws + 2 * (size_t)NT };

    const int n4T       = NT / 4;                          // 3.2M float4
    const int initGrid  = (n4T + 255) / 256;
    const int n4O       = (NROWS * FOUT) / 4;
    const int outGrid   = (n4O + 255) / 256;
    const int edgeGrid  = (EDG + 3) / 4;
    const int gemmGrid  = (NROWS / 16 + 7) / 8;            // 12500 tiles / 8 waves

    // out = 0, then accumulate 6 GEMM chunks.
    init_scale_kernel<<<outGrid, 256, 0, stream>>>((float4*)out, nullptr, 0.f, n4O);

    // k = 0: A = x directly.
    gemm_wmma_chunk_kernel<<<gemmGrid, 256, 0, stream>>>(out, x, W, 0, NROWS);

    // k = 1: T1 = L x
    init_scale_kernel<<<initGrid, 256, 0, stream>>>((float4*)T[0], nullptr, 0.f, n4T);
    cheb_spmm_edges_kernel<<<edgeGrid, 256, 0, stream>>>(T[0], x, vals, row, col, 1.0f, EDG);
    gemm_wmma_chunk_kernel<<<gemmGrid, 256, 0, stream>>>(out, T[0], W, 1, NROWS);

    // k >= 2: T_k = 2 L T_{k-1} - T_{k-2}, rotating the 3 buffers.
    const float* prev2 = x;     // T_{k-2}
    float*       prev1 = T[0];  // T_{k-1}
    for (int k = 2; k < KCHEB; ++k) {
        float* cur = T[(k - 1) % 3];
        init_scale_kernel<<<initGrid, 256, 0, stream>>>(
            (float4*)cur, (const float4*)prev2, -1.0f, n4T);
        cheb_spmm_edges_kernel<<<edgeGrid, 256, 0, stream>>>(
            cur, prev1, vals, row, col, 2.0f, EDG);
        gemm_wmma_chunk_kernel<<<gemmGrid, 256, 0, stream>>>(out, cur, W, k, NROWS);
        prev2 = prev1;
        prev1 = cur;
    }
}